// LaplacianModule_30511447671635
// MI455X (gfx1250) — compile-verified
//
#include <hip/hip_runtime.h>

// Problem constants from the reference: B, N, FN = (16, 100000, 200000)
#define B_   16
#define N_   100000
#define FN_  200000

// ---------------------------------------------------------------------------
// Zero-init the output (harness poisons d_out; segment_sum semantics need 0).
// float4 stores: 19.2 MB -> ~1 us at HBM rate.
// ---------------------------------------------------------------------------
__global__ void __launch_bounds__(256)
lap_zero_kernel(float* __restrict__ out, long long n) {
  const long long i4 = (((long long)blockIdx.x * blockDim.x) + threadIdx.x) * 4;
  if (i4 + 4 <= n) {
    float4 z; z.x = 0.0f; z.y = 0.0f; z.z = 0.0f; z.w = 0.0f;
    *reinterpret_cast<float4*>(out + i4) = z;
  } else if (i4 < n) {
    for (long long i = i4; i < n; ++i) out[i] = 0.0f;
  }
}

// Native f32 add atomic (global_atomic_add_f32), no CAS fallback.
__device__ __forceinline__ void atomAddF32(float* p, float v) {
  unsafeAtomicAdd(p, v);
}

// ---------------------------------------------------------------------------
// Fused cotangent-weight + symmetric scatter kernel. One thread per face.
//
// Reference mapping (0-indexed verts v0,v1,v2 = ref v1,v2,v3):
//   l1 = |v1-v2|, l2 = |v2-v0|, l3 = |v0-v1|
//   sp = (l1+l2+l3)/2 ; A = 2*sqrt(sp(sp-l1)(sp-l2)(sp-l3))
//   w0 = (l2^2+l3^2-l1^2)/(4A)   edge (F1,F2)
//   w1 = (l1^2+l3^2-l2^2)/(4A)   edge (F2,F0)
//   w2 = (l1^2+l2^2-l3^2)/(4A)   edge (F0,F1)
// Each edge (r,c,w) contributes w*(V[c]-V[r]) to out[r] and the negation to
// out[c]; folding per face gives one 3-vector per vertex -> 9 atomics/face.
// ---------------------------------------------------------------------------
__global__ void __launch_bounds__(256)
lap_cot_scatter_kernel(const float* __restrict__ V,
                       const int*   __restrict__ F,
                       float*       __restrict__ out) {
  const int f = blockIdx.x * blockDim.x + threadIdx.x;
  const int b = blockIdx.y;
  if (f >= FN_) return;

  const int* __restrict__ Fp = F + ((long long)b * FN_ + (long long)f) * 3;
  const int i0 = Fp[0];
  const int i1 = Fp[1];
  const int i2 = Fp[2];

  const float* __restrict__ Vb = V + (long long)b * (long long)(N_ * 3);
  const float v0x = Vb[3 * i0 + 0], v0y = Vb[3 * i0 + 1], v0z = Vb[3 * i0 + 2];
  const float v1x = Vb[3 * i1 + 0], v1y = Vb[3 * i1 + 1], v1z = Vb[3 * i1 + 2];
  const float v2x = Vb[3 * i2 + 0], v2y = Vb[3 * i2 + 1], v2z = Vb[3 * i2 + 2];

  // Edge vectors and squared lengths
  const float d1x = v1x - v2x, d1y = v1y - v2y, d1z = v1z - v2z;
  const float d2x = v2x - v0x, d2y = v2y - v0y, d2z = v2z - v0z;
  const float d3x = v0x - v1x, d3y = v0y - v1y, d3z = v0z - v1z;
  const float l1s = d1x * d1x + d1y * d1y + d1z * d1z;
  const float l2s = d2x * d2x + d2y * d2y + d2z * d2z;
  const float l3s = d3x * d3x + d3y * d3y + d3z * d3z;
  const float l1 = sqrtf(l1s);
  const float l2 = sqrtf(l2s);
  const float l3 = sqrtf(l3s);

  // Heron: A = 2*sqrt(sp(sp-l1)(sp-l2)(sp-l3)); weights are cot/(4A)
  const float sp    = 0.5f * (l1 + l2 + l3);
  const float heron = sp * (sp - l1) * (sp - l2) * (sp - l3);
  const float inv4A = 0.125f / sqrtf(heron);   // 1/(4A) = 0.125/sqrt(heron)

  const float w0 = (l2s + l3s - l1s) * inv4A;
  const float w1 = (l1s + l3s - l2s) * inv4A;
  const float w2 = (l1s + l2s - l3s) * inv4A;

  // Folded per-vertex gradients:
  //   g0 = w1*(v2-v0) + w2*(v1-v0)
  //   g1 = w0*(v2-v1) + w2*(v0-v1)
  //   g2 = w0*(v1-v2) + w1*(v0-v2)
  const float g0x = w1 * (v2x - v0x) + w2 * (v1x - v0x);
  const float g0y = w1 * (v2y - v0y) + w2 * (v1y - v0y);
  const float g0z = w1 * (v2z - v0z) + w2 * (v1z - v0z);

  const float g1x = w0 * (v2x - v1x) + w2 * (v0x - v1x);
  const float g1y = w0 * (v2y - v1y) + w2 * (v0y - v1y);
  const float g1z = w0 * (v2z - v1z) + w2 * (v0z - v1z);

  const float g2x = w0 * (v1x - v2x) + w1 * (v0x - v2x);
  const float g2y = w0 * (v1y - v2y) + w1 * (v0y - v2y);
  const float g2z = w0 * (v1z - v2z) + w1 * (v0z - v2z);

  float* __restrict__ ob = out + (long long)b * (long long)(N_ * 3);
  atomAddF32(&ob[3 * i0 + 0], g0x);
  atomAddF32(&ob[3 * i0 + 1], g0y);
  atomAddF32(&ob[3 * i0 + 2], g0z);
  atomAddF32(&ob[3 * i1 + 0], g1x);
  atomAddF32(&ob[3 * i1 + 1], g1y);
  atomAddF32(&ob[3 * i1 + 2], g1z);
  atomAddF32(&ob[3 * i2 + 0], g2x);
  atomAddF32(&ob[3 * i2 + 1], g2y);
  atomAddF32(&ob[3 * i2 + 2], g2z);
}

extern "C" void kernel_launch(void* const* d_in, const int* in_sizes, int n_in,
                              void* d_out, int out_size, void* d_ws, size_t ws_size,
                              hipStream_t stream) {
  (void)in_sizes; (void)n_in; (void)d_ws; (void)ws_size;

  const float* V = (const float*)d_in[0];   // (B, N, 3) f32
  const int*   F = (const int*)d_in[1];     // (B, FN, 3) i32
  float*       out = (float*)d_out;         // (B, N, 3) f32

  // 1) zero output
  const long long n  = (long long)out_size;           // 4,800,000
  const long long n4 = (n + 3) / 4;
  const int zblocks  = (int)((n4 + 255) / 256);
  lap_zero_kernel<<<zblocks, 256, 0, stream>>>(out, n);

  // 2) fused cotangent + scatter, one thread per face
  dim3 block(256, 1, 1);
  dim3 grid((FN_ + 255) / 256, B_, 1);
  lap_cot_scatter_kernel<<<grid, block, 0, stream>>>(V, F, out);
}